// MaximumLikelihoodDetector_64828236366362
// MI455X (gfx1250) — compile-verified
//
#include <hip/hip_runtime.h>
#include <math.h>

typedef __attribute__((ext_vector_type(2))) float v2f;
typedef __attribute__((ext_vector_type(8))) float v8f;

#define BIGF 1e30f
#define NBATCH 4096
#define UD 24   // reduced inner dimension

// ---------------------------------------------------------------------------
// Phase 0: candidate table v[c][24]
//   v = [Re P (9, k*3+l), -Im P (9), -2*Re x (3), +2*Im x (3)]
//   P[k,l] = conj(x_k) * x_l
// ---------------------------------------------------------------------------
__device__ __forceinline__ void qam16_point(int s, float& re, float& im) {
    // Gray-coded 16-QAM, bits (b0 b1 b2 b3), real from (b0,b2), imag from (b1,b3)
    const int b0 = (s >> 3) & 1, b1 = (s >> 2) & 1, b2 = (s >> 1) & 1, b3 = s & 1;
    const float inv = 0.3162277660168379f; // 1/sqrt(10)
    re = (float)((1 - 2 * b0) * (2 - (1 - 2 * b2))) * inv;
    im = (float)((1 - 2 * b1) * (2 - (1 - 2 * b3))) * inv;
}

__global__ void build_v_kernel(float* __restrict__ v) {
    const int c = blockIdx.x * blockDim.x + threadIdx.x;
    if (c >= 4096) return;
    const int idx0 = (c >> 8) & 15, idx1 = (c >> 4) & 15, idx2 = c & 15;
    float xr[3], xi[3];
    qam16_point(idx0, xr[0], xi[0]);
    qam16_point(idx1, xr[1], xi[1]);
    qam16_point(idx2, xr[2], xi[2]);
    float vv[UD];
#pragma unroll
    for (int k = 0; k < 3; k++) {
#pragma unroll
        for (int l = 0; l < 3; l++) {
            const float pre = xr[k] * xr[l] + xi[k] * xi[l];
            const float pim = xr[k] * xi[l] - xi[k] * xr[l];
            vv[k * 3 + l]     = pre;
            vv[9 + k * 3 + l] = -pim;
        }
    }
#pragma unroll
    for (int k = 0; k < 3; k++) {
        vv[18 + k] = -2.0f * xr[k];
        vv[21 + k] =  2.0f * xi[k];
    }
#pragma unroll
    for (int t = 0; t < UD; t++) v[c * UD + t] = vv[t];
}

// ---------------------------------------------------------------------------
// Phase 1: per-batch whitening -> u[b][24]
//   L = chol(S) (complex, lower); yw = L^-1 y; hw = L^-1 h
//   G = hw^H hw (3x3), z = yw^H hw (3)
//   u = [Re G (9), Im G (9), Re z (3), Im z (3)]
// ---------------------------------------------------------------------------
__global__ void build_u_kernel(const float* __restrict__ y_re, const float* __restrict__ y_im,
                               const float* __restrict__ h_re, const float* __restrict__ h_im,
                               const float* __restrict__ s_re, const float* __restrict__ s_im,
                               float* __restrict__ u) {
    const int b = blockIdx.x * blockDim.x + threadIdx.x;
    if (b >= NBATCH) return;

    float Lr[8][8], Li[8][8];
#pragma unroll
    for (int i = 0; i < 8; i++) {
#pragma unroll
        for (int j = 0; j <= i; j++) {
            Lr[i][j] = s_re[b * 64 + i * 8 + j];
            Li[i][j] = s_im[b * 64 + i * 8 + j];
        }
    }
    // complex Cholesky (lower), fully unrolled
#pragma unroll
    for (int j = 0; j < 8; j++) {
        float d = Lr[j][j];
#pragma unroll
        for (int p = 0; p < j; p++) d -= Lr[j][p] * Lr[j][p] + Li[j][p] * Li[j][p];
        const float dj = sqrtf(d);
        const float inv = 1.0f / dj;
        Lr[j][j] = dj; Li[j][j] = 0.0f;
#pragma unroll
        for (int i = j + 1; i < 8; i++) {
            float ar = Lr[i][j], ai = Li[i][j];
#pragma unroll
            for (int p = 0; p < j; p++) {
                // subtract L[i][p] * conj(L[j][p])
                ar -= Lr[i][p] * Lr[j][p] + Li[i][p] * Li[j][p];
                ai -= Li[i][p] * Lr[j][p] - Lr[i][p] * Li[j][p];
            }
            Lr[i][j] = ar * inv; Li[i][j] = ai * inv;
        }
    }
    // forward solve: yw
    float ywr[8], ywi[8];
#pragma unroll
    for (int i = 0; i < 8; i++) {
        float ar = y_re[b * 8 + i], ai = y_im[b * 8 + i];
#pragma unroll
        for (int j = 0; j < i; j++) {
            ar -= Lr[i][j] * ywr[j] - Li[i][j] * ywi[j];
            ai -= Lr[i][j] * ywi[j] + Li[i][j] * ywr[j];
        }
        const float inv = 1.0f / Lr[i][i];
        ywr[i] = ar * inv; ywi[i] = ai * inv;
    }
    // forward solve: hw (3 columns), h shape (B, M=8, K=3)
    float hwr[8][3], hwi[8][3];
#pragma unroll
    for (int k = 0; k < 3; k++) {
#pragma unroll
        for (int i = 0; i < 8; i++) {
            float ar = h_re[b * 24 + i * 3 + k], ai = h_im[b * 24 + i * 3 + k];
#pragma unroll
            for (int j = 0; j < i; j++) {
                ar -= Lr[i][j] * hwr[j][k] - Li[i][j] * hwi[j][k];
                ai -= Lr[i][j] * hwi[j][k] + Li[i][j] * hwr[j][k];
            }
            const float inv = 1.0f / Lr[i][i];
            hwr[i][k] = ar * inv; hwi[i][k] = ai * inv;
        }
    }
    float uu[UD];
#pragma unroll
    for (int k = 0; k < 3; k++) {
#pragma unroll
        for (int l = 0; l < 3; l++) {
            float gr = 0.f, gi = 0.f;
#pragma unroll
            for (int m = 0; m < 8; m++) {
                gr += hwr[m][k] * hwr[m][l] + hwi[m][k] * hwi[m][l];
                gi += hwr[m][k] * hwi[m][l] - hwi[m][k] * hwr[m][l];
            }
            uu[k * 3 + l]     = gr;
            uu[9 + k * 3 + l] = gi;
        }
    }
#pragma unroll
    for (int k = 0; k < 3; k++) {
        float zr = 0.f, zi = 0.f;
#pragma unroll
        for (int m = 0; m < 8; m++) {
            zr += ywr[m] * hwr[m][k] + ywi[m] * hwi[m][k];
            zi += ywr[m] * hwi[m][k] - ywi[m] * hwr[m][k];
        }
        uu[18 + k] = zr;
        uu[21 + k] = zi;
    }
#pragma unroll
    for (int t = 0; t < UD; t++) u[b * UD + t] = uu[t];
}

// ---------------------------------------------------------------------------
// Phase 2: fused WMMA GEMM (cand x batch, K=24 via 6x V_WMMA_F32_16X16X4_F32)
//          + hierarchical masked-min reduction -> LLRs
// Block: 256 threads (8 waves); block owns 16 batches; wave w owns i0 in {2w,2w+1}
// D-tile layout (f32 16x16): lane l -> batch N = l&15; slot r -> cand M = r + 8*(l>>4)
// ---------------------------------------------------------------------------
__global__ void __launch_bounds__(256)
detect_kernel(const float* __restrict__ u, const float* __restrict__ v,
              float* __restrict__ out) {
    __shared__ float lm1[8][32][16]; // per-wave, per-lane partial min over (i0 pair, i2-half) for each i1
    __shared__ float lm2[8][32][8];  // per-wave, per-lane partial min over (i0 pair, all i1) for i2 = r + 8*half
    __shared__ float lb0[8][32][2];  // per-wave, per-lane min over (all i1, i2-half) for each of its 2 i0

    const int tid  = threadIdx.x;
    const int wave = tid >> 5;
    const int lane = tid & 31;
    const int li   = lane & 15;
    const int hh   = lane >> 4;
    const int bbase = blockIdx.x * 16;

    // B-matrix fragments (batch u-vectors), constant across all tiles of this wave.
    // f32 4x16 fragment: vgpr0 = K(base+2h), vgpr1 = K(base+2h+1) -> contiguous float2.
    const float* ub = u + (size_t)(bbase + li) * UD + 2 * hh;
    v2f bf[6];
#pragma unroll
    for (int kk = 0; kk < 6; kk++) bf[kk] = *(const v2f*)(ub + 4 * kk);

    float m1h[16], m2h[8], bm0[2];
#pragma unroll
    for (int t = 0; t < 16; t++) m1h[t] = BIGF;
#pragma unroll
    for (int t = 0; t < 8; t++) m2h[t] = BIGF;

#pragma unroll
    for (int j = 0; j < 2; j++) {
        const int i0 = wave * 2 + j;
        float bmin = BIGF;
#pragma unroll
        for (int i1 = 0; i1 < 16; i1++) {
            const int cbase = i0 * 256 + i1 * 16;
            const float* vp = v + (size_t)(cbase + li) * UD + 2 * hh;
            v8f acc = {0.f, 0.f, 0.f, 0.f, 0.f, 0.f, 0.f, 0.f};
#pragma unroll
            for (int kk = 0; kk < 6; kk++) {
                const v2f af = *(const v2f*)(vp + 4 * kk);
                acc = __builtin_amdgcn_wmma_f32_16x16x4_f32(
                    false, af, false, bf[kk], (short)0, acc, false, false);
            }
            const float tmin =
                fminf(fminf(fminf(acc[0], acc[1]), fminf(acc[2], acc[3])),
                      fminf(fminf(acc[4], acc[5]), fminf(acc[6], acc[7])));
            bmin = fminf(bmin, tmin);
            m1h[i1] = fminf(m1h[i1], tmin);
#pragma unroll
            for (int r = 0; r < 8; r++) m2h[r] = fminf(m2h[r], acc[r]);
        }
        bm0[j] = bmin;
    }

#pragma unroll
    for (int t = 0; t < 16; t++) lm1[wave][lane][t] = m1h[t];
#pragma unroll
    for (int t = 0; t < 8; t++) lm2[wave][lane][t] = m2h[t];
    lb0[wave][lane][0] = bm0[0];
    lb0[wave][lane][1] = bm0[1];
    __syncthreads();

    // Final assembly: 16 lanes, one batch each.
    if (tid < 16) {
        const int b = tid;
        float m[3][16];
#pragma unroll
        for (int i0 = 0; i0 < 16; i0++) {
            const int w = i0 >> 1, jj = i0 & 1;
            m[0][i0] = fminf(lb0[w][b][jj], lb0[w][b + 16][jj]);
        }
#pragma unroll
        for (int i1 = 0; i1 < 16; i1++) {
            float t = BIGF;
#pragma unroll
            for (int w = 0; w < 8; w++)
                t = fminf(t, fminf(lm1[w][b][i1], lm1[w][b + 16][i1]));
            m[1][i1] = t;
        }
#pragma unroll
        for (int i2 = 0; i2 < 16; i2++) {
            const int h2 = i2 >> 3, r = i2 & 7;
            float t = BIGF;
#pragma unroll
            for (int w = 0; w < 8; w++) t = fminf(t, lm2[w][b + 16 * h2][r]);
            m[2][i2] = t;
        }
        // LLRs: bit n of symbol k is (idx >> (3-n)) & 1 ; llr = min0 - min1
#pragma unroll
        for (int k = 0; k < 3; k++) {
#pragma unroll
            for (int n = 0; n < 4; n++) {
                float mn0 = BIGF, mn1 = BIGF;
#pragma unroll
                for (int i = 0; i < 16; i++) {
                    if ((i >> (3 - n)) & 1) mn1 = fminf(mn1, m[k][i]);
                    else                    mn0 = fminf(mn0, m[k][i]);
                }
                out[(size_t)(bbase + b) * 12 + k * 4 + n] = mn0 - mn1;
            }
        }
    }
}

// ---------------------------------------------------------------------------
extern "C" void kernel_launch(void* const* d_in, const int* in_sizes, int n_in,
                              void* d_out, int out_size, void* d_ws, size_t ws_size,
                              hipStream_t stream) {
    const float* y_re = (const float*)d_in[0];
    const float* y_im = (const float*)d_in[1];
    const float* h_re = (const float*)d_in[2];
    const float* h_im = (const float*)d_in[3];
    const float* s_re = (const float*)d_in[4];
    const float* s_im = (const float*)d_in[5];
    float* out = (float*)d_out;

    float* u_tab = (float*)d_ws;                 // 4096*24 floats
    float* v_tab = u_tab + (size_t)NBATCH * UD;  // 4096*24 floats

    build_v_kernel<<<16, 256, 0, stream>>>(v_tab);
    build_u_kernel<<<16, 256, 0, stream>>>(y_re, y_im, h_re, h_im, s_re, s_im, u_tab);
    detect_kernel<<<NBATCH / 16, 256, 0, stream>>>(u_tab, v_tab, out);
}